// GraphSAGELayer_25305947308264
// MI455X (gfx1250) — compile-verified
//
#include <hip/hip_runtime.h>
#include <hip/hip_bf16.h>

typedef __attribute__((ext_vector_type(2))) float v2f;
typedef __attribute__((ext_vector_type(8))) float v8f;

#define B_ 8
#define N_ 5000
#define E_ 100000
#define D_ 128
#define LD_ 132   // padded LDS row stride (floats): bank = (4*row + col) % 64 -> conflict-free

// ---------------------------------------------------------------------------
// Kernel 0: zero the workspace (sums + counts)
// ---------------------------------------------------------------------------
__global__ void zero_ws_kernel(float* __restrict__ p, long n) {
    long i = (long)blockIdx.x * blockDim.x + threadIdx.x;
    if (i < n) p[i] = 0.0f;
}

// ---------------------------------------------------------------------------
// Kernel 0b: pack weights into K-pair (v2f) layout so the GEMM hot loop can
// fetch both K-values of a WMMA operand with one global_load_b64.
//   Wp[p*D + n] = { W[2p][n], W[2p+1][n] },  p = 0..63
// ---------------------------------------------------------------------------
__global__ void pack_w_kernel(const float* __restrict__ Ws,
                              const float* __restrict__ Wn,
                              v2f* __restrict__ Wsp, v2f* __restrict__ Wnp) {
    int i = blockIdx.x * blockDim.x + threadIdx.x;
    if (i < (D_ / 2) * D_) {
        int p = i >> 7;          // K-pair index
        int n = i & (D_ - 1);    // output column
        v2f a, b;
        a.x = Ws[(2 * p) * D_ + n];
        a.y = Ws[(2 * p + 1) * D_ + n];
        b.x = Wn[(2 * p) * D_ + n];
        b.y = Wn[(2 * p + 1) * D_ + n];
        Wsp[i] = a;
        Wnp[i] = b;
    }
}

// ---------------------------------------------------------------------------
// Kernel 1: edge scatter-add (masked segment sum). One wave32 per edge;
// one float4 (global_load_b128) per lane covers D=128, atomics resolve in L2
// (working set ~21 MB << 192 MB L2).
// ---------------------------------------------------------------------------
__global__ __launch_bounds__(256) void edge_scatter_kernel(
    const float* __restrict__ x,      // [B,N,D]
    const int*   __restrict__ ei,     // [B,2,E]
    const float* __restrict__ em,     // [B,E]
    float* __restrict__ sums,         // [B,N,D]
    float* __restrict__ cnt)          // [B,N]
{
    int gwave = (blockIdx.x * blockDim.x + threadIdx.x) >> 5;
    int lane  = threadIdx.x & 31;
    long total = (long)B_ * E_;
    if (gwave >= total) return;
    int b = gwave / E_;
    int e = gwave - b * E_;

    const int* eib = ei + (long)b * 2 * E_;
    int src = eib[e];
    int tgt = eib[E_ + e];
    float m = em[(long)b * E_ + e];
    if (m != 0.0f) {
        const float4* xs4 = reinterpret_cast<const float4*>(
            x + ((long)b * N_ + src) * D_);
        float* sd = sums + ((long)b * N_ + tgt) * D_ + 4 * lane;
        float4 v = xs4[lane];                 // 16B aligned: row base 512B, off 16B*lane
        atomicAdd(&sd[0], v.x * m);
        atomicAdd(&sd[1], v.y * m);
        atomicAdd(&sd[2], v.z * m);
        atomicAdd(&sd[3], v.w * m);
        if (lane == 0) atomicAdd(&cnt[(long)b * N_ + tgt], m);
    }
}

// ---------------------------------------------------------------------------
// Kernel 2: fused node update.
//   out = LayerNorm(relu(x@W_self + b_self + (sums/max(cnt,1))@W_nb + b_nb))
//         * gamma + beta, masked.
// Block = 256 threads (8 waves) handles 16 nodes. Wave w computes output
// columns [16w,16w+16) via V_WMMA_F32_16X16X4_F32 (full fp32), accumulating
// both GEMMs into one f32 C/D register block.
// ---------------------------------------------------------------------------
__global__ __launch_bounds__(256) void node_update_kernel(
    const float* __restrict__ x,      // [B,N,D]
    const float* __restrict__ nmask,  // [B,N]
    const v2f*   __restrict__ Wsp,    // [D/2, D] packed K-pairs (self)
    const float* __restrict__ bs,     // [D]
    const v2f*   __restrict__ Wnp,    // [D/2, D] packed K-pairs (neighbor)
    const float* __restrict__ bn,     // [D]
    const float* __restrict__ gamma,  // [D]
    const float* __restrict__ beta,   // [D]
    const float* __restrict__ sums,   // [B,N,D]
    const float* __restrict__ cnt,    // [B,N]
    float* __restrict__ out,          // [B,N,D]
    int ntiles)
{
    __shared__ float sA[16 * LD_];   // self features tile
    __shared__ float sG[16 * LD_];   // aggregated features tile
    __shared__ float sO[16 * LD_];   // pre-norm output tile

    int tid  = threadIdx.x;
    int tile = blockIdx.x % ntiles;
    int b    = blockIdx.x / ntiles;
    int row0 = tile * 16;

    // Cooperative tile load: 16 nodes x 128 dims for both A matrices.
    for (int i = tid; i < 16 * D_; i += 256) {
        int r = i >> 7;
        int c = i & (D_ - 1);
        int node = row0 + r;
        float a = 0.0f, g = 0.0f;
        if (node < N_) {
            long base = ((long)b * N_ + node) * D_ + c;
            a = x[base];
            float ct = cnt[(long)b * N_ + node];
            g = sums[base] / fmaxf(ct, 1.0f);
        }
        sA[r * LD_ + c] = a;
        sG[r * LD_ + c] = g;
    }
    __syncthreads();

    int wave = tid >> 5;          // 0..7 -> output column tile
    int lane = tid & 31;
    int ln   = lane & 15;         // position within 16-lane half
    int hi   = lane >> 4;         // 0 or 1
    int n    = wave * 16 + ln;    // output column handled by this lane

    // WMMA f32 16x16x4: A 16x4 (lanes 0-15: K=0/1, lanes 16-31: K=2/3),
    // B 4x16 (mirrored), C/D 16x16 (VGPR r: M=r for lanes 0-15, M=r+8 else).
    v8f acc = {};
    for (int k0 = 0; k0 < D_; k0 += 4) {
        v2f a;
        a.x = sA[ln * LD_ + k0 + 2 * hi + 0];
        a.y = sA[ln * LD_ + k0 + 2 * hi + 1];
        v2f w = Wsp[((k0 >> 1) + hi) * D_ + n];   // one global_load_b64
        acc = __builtin_amdgcn_wmma_f32_16x16x4_f32(
            false, a, false, w, (short)0, acc, false, false);
    }
    for (int k0 = 0; k0 < D_; k0 += 4) {
        v2f a;
        a.x = sG[ln * LD_ + k0 + 2 * hi + 0];
        a.y = sG[ln * LD_ + k0 + 2 * hi + 1];
        v2f w = Wnp[((k0 >> 1) + hi) * D_ + n];
        acc = __builtin_amdgcn_wmma_f32_16x16x4_f32(
            false, a, false, w, (short)0, acc, false, false);
    }

    // bias + relu -> LDS (pre-norm activations)
    float bias = bs[n] + bn[n];
#pragma unroll
    for (int r = 0; r < 8; ++r) {
        int row = r + 8 * hi;
        float v = fmaxf(acc[r] + bias, 0.0f);
        sO[row * LD_ + n] = v;
    }
    __syncthreads();

    // LayerNorm: wave w handles node rows 2w (lanes 0-15) and 2w+1 (16-31).
    // Each lane owns 8 columns; 16-lane xor-shuffle reduction (offsets 1..8
    // flip only bits 0-3 -> stay within each 16-lane half of the wave32).
    int lrow = wave * 2 + hi;     // 0..15
    float vals[8];
    float s1 = 0.0f, s2 = 0.0f;
#pragma unroll
    for (int j = 0; j < 8; ++j) {
        float v = sO[lrow * LD_ + ln + 16 * j];
        vals[j] = v;
        s1 += v;
        s2 += v * v;
    }
#pragma unroll
    for (int off = 1; off < 16; off <<= 1) {
        s1 += __shfl_xor(s1, off, 32);
        s2 += __shfl_xor(s2, off, 32);
    }
    const float inv_d = 1.0f / (float)D_;
    float mu   = s1 * inv_d;
    float var  = s2 * inv_d - mu * mu;
    float rsig = rsqrtf(var + 1e-5f);

    int node = row0 + lrow;
    if (node < N_) {
        float msk  = nmask[(long)b * N_ + node];
        long obase = ((long)b * N_ + node) * D_;
#pragma unroll
        for (int j = 0; j < 8; ++j) {
            int c = ln + 16 * j;
            float v = (vals[j] - mu) * rsig * gamma[c] + beta[c];
            out[obase + c] = v * msk;
        }
    }
}

// ---------------------------------------------------------------------------
extern "C" void kernel_launch(void* const* d_in, const int* in_sizes, int n_in,
                              void* d_out, int out_size, void* d_ws, size_t ws_size,
                              hipStream_t stream) {
    const float* x     = (const float*)d_in[0];   // node_features [B,N,D]
    const int*   ei    = (const int*)  d_in[1];   // edge_index    [B,2,E]
    const float* nmask = (const float*)d_in[2];   // node_mask     [B,N]
    const float* em    = (const float*)d_in[3];   // edge_mask     [B,E]
    const float* Ws    = (const float*)d_in[4];   // W_self [D,D]
    const float* bs    = (const float*)d_in[5];   // b_self [D]
    const float* Wn    = (const float*)d_in[6];   // W_nb   [D,D]
    const float* bn    = (const float*)d_in[7];   // b_nb   [D]
    const float* gamma = (const float*)d_in[8];   // [D]
    const float* beta  = (const float*)d_in[9];   // [D]
    float* out = (float*)d_out;

    // Workspace layout (floats): sums [B*N*D] | cnt [B*N] | Wsp [D/2*D*2] | Wnp [D/2*D*2]
    float* sums = (float*)d_ws;
    float* cnt  = sums + (long)B_ * N_ * D_;
    float* wsbase = cnt + (long)B_ * N_;              // 8-byte aligned (20,640,000 B)
    v2f* Wsp = (v2f*)wsbase;                          // 8192 v2f = 64 KB
    v2f* Wnp = Wsp + (long)(D_ / 2) * D_;             // 64 KB
    long ws_elems = (long)B_ * N_ * D_ + (long)B_ * N_;

    // 1) zero atomics workspace
    {
        long n = ws_elems;
        int blocks = (int)((n + 255) / 256);
        zero_ws_kernel<<<blocks, 256, 0, stream>>>(sums, n);
    }
    // 1b) pack weights into K-pair layout for b64 operand loads
    {
        int n = (D_ / 2) * D_;
        pack_w_kernel<<<(n + 255) / 256, 256, 0, stream>>>(Ws, Wn, Wsp, Wnp);
    }
    // 2) edge scatter (one wave per edge)
    {
        long waves = (long)B_ * E_;
        int blocks = (int)((waves * 32 + 255) / 256);
        edge_scatter_kernel<<<blocks, 256, 0, stream>>>(x, ei, em, sums, cnt);
    }
    // 3) fused GEMM + relu + layernorm + mask
    {
        int ntiles = (N_ + 15) / 16;
        int blocks = B_ * ntiles;
        node_update_kernel<<<blocks, 256, 0, stream>>>(
            x, nmask, Wsp, bs, Wnp, bn, gamma, beta, sums, cnt, out, ntiles);
    }
}